// NeuralSigKer_5789615915575
// MI455X (gfx1250) — compile-verified
//
#include <hip/hip_runtime.h>
#include <hip/hip_bf16.h>

// Neural signature kernel PDE on gfx1250 (MI455X).
// K[s+1,t] = K[s,t] + exclusive_prefix_t( (vA*K[s,j]+vb) * <dX_s, dY_j> )
//
// Pipeline per batch (1 block = 16 waves):
//   waves 1..15: V_WMMA_F32_16X16X4_F32 produce next 16-row band of
//                P = dX @ dY^T into double-buffered LDS
//   wave 0:      consumes current band: 16 recurrence steps, each a
//                barrier-free register scan (lane owns 16 contiguous t's)
// One __syncthreads per band (16 steps), WMMA fully hidden behind the
// serial scan chain, which is the true limiter (output = 128 MB => ~6us
// HBM floor; the 511-step dependency chain dominates).

typedef __attribute__((ext_vector_type(2))) float v2f;
typedef __attribute__((ext_vector_type(8))) float v8f;

#define NSK_B 128
#define NSK_T 512
#define NSK_D 16

#define VAR0 1.0f   // SIGMA_0^2
#define VARA 1.0f   // SIGMA_A^2
#define VARB 0.25f  // SIGMA_B^2

__global__ __launch_bounds__(512) void nsk_kernel(const float* __restrict__ X,
                                                  const float* __restrict__ Y,
                                                  float* __restrict__ out) {
  __shared__ float sDX[NSK_T * NSK_D];     // 32 KB, row 511 zero-padded
  __shared__ float sDY[NSK_T * NSK_D];     // 32 KB, row 511 zero-padded
  __shared__ float sP[2][16 * NSK_T];      // 2 x 32 KB band double-buffer

  const int b    = blockIdx.x;
  const int tid  = threadIdx.x;  // 0..511
  const int lane = tid & 31;
  const int wid  = tid >> 5;     // 0..15

  const float* Xb = X + (size_t)b * NSK_T * NSK_D;
  const float* Yb = Y + (size_t)b * NSK_T * NSK_D;
  float* Ob = out + (size_t)b * NSK_T * NSK_T;

  // ---- build increment rows dX, dY in LDS (one 16-wide row per thread) ----
  if (tid < NSK_T - 1) {
#pragma unroll
    for (int d0 = 0; d0 < NSK_D; d0 += 4) {
      float4 x0 = *(const float4*)(Xb + tid * NSK_D + d0);
      float4 x1 = *(const float4*)(Xb + (tid + 1) * NSK_D + d0);
      sDX[tid * NSK_D + d0 + 0] = x1.x - x0.x;
      sDX[tid * NSK_D + d0 + 1] = x1.y - x0.y;
      sDX[tid * NSK_D + d0 + 2] = x1.z - x0.z;
      sDX[tid * NSK_D + d0 + 3] = x1.w - x0.w;
      float4 y0 = *(const float4*)(Yb + tid * NSK_D + d0);
      float4 y1 = *(const float4*)(Yb + (tid + 1) * NSK_D + d0);
      sDY[tid * NSK_D + d0 + 0] = y1.x - y0.x;
      sDY[tid * NSK_D + d0 + 1] = y1.y - y0.y;
      sDY[tid * NSK_D + d0 + 2] = y1.z - y0.z;
      sDY[tid * NSK_D + d0 + 3] = y1.w - y0.w;
    }
  } else {
#pragma unroll
    for (int d0 = 0; d0 < NSK_D; ++d0) {
      sDX[tid * NSK_D + d0] = 0.0f;   // dX row 511 = 0 (pads band 31)
      sDY[tid * NSK_D + d0] = 0.0f;   // dY row 511 = 0 => P[:,511] = 0 => g[511] = 0
    }
  }
  __syncthreads();

  const int nhalf = lane & 15;  // A/C row (M) or B/C column (N)
  const int kh    = lane >> 4;  // lane-half selector per ISA fragment layout

  // A 16x4 f32 frag: VGPR v of lane L holds A[M = L&15][K = v + 2*(L>>4)]
  // B 4x16 f32 frag: VGPR v of lane L holds B[K = v + 2*(L>>4)][N = L&15]
  // C 16x16 f32:     VGPR v of lane L holds C[M = v + 8*(L>>4)][N = L&15]
  auto wmma_band = [&](int sb, int buf, int tstart, int tstep) {
    for (int tt = tstart; tt < 32; tt += tstep) {
      v8f c = {0.f, 0.f, 0.f, 0.f, 0.f, 0.f, 0.f, 0.f};
#pragma unroll
      for (int k4 = 0; k4 < NSK_D; k4 += 4) {
        const int ka = k4 + 2 * kh;
        v2f a, bb;
        a.x  = sDX[(sb + nhalf) * NSK_D + ka];
        a.y  = sDX[(sb + nhalf) * NSK_D + ka + 1];
        bb.x = sDY[(tt * 16 + nhalf) * NSK_D + ka];
        bb.y = sDY[(tt * 16 + nhalf) * NSK_D + ka + 1];
        c = __builtin_amdgcn_wmma_f32_16x16x4_f32(
            /*neg_a=*/false, a, /*neg_b=*/false, bb,
            /*c_mod=*/(short)0, c, /*reuse_a=*/false, /*reuse_b=*/false);
      }
#pragma unroll
      for (int v = 0; v < 8; ++v)
        sP[buf][(v + 8 * kh) * NSK_T + tt * 16 + nhalf] = c[v];
    }
  };

  // ---- prologue: all 16 waves build band 0 into buffer 0 ----
  wmma_band(0, 0, wid, 16);
  __syncthreads();

  // ---- wave 0 state: lane L owns K-row elements t = 16L .. 16L+15 ----
  float prev[16];
  if (wid == 0) {
#pragma unroll
    for (int i = 0; i < 16; ++i) prev[i] = VAR0;
    const float4 r0 = {VAR0, VAR0, VAR0, VAR0};
#pragma unroll
    for (int i = 0; i < 16; i += 4)
      *(float4*)(Ob + lane * 16 + i) = r0;   // K[b,0,:] = var0
  }

  for (int band = 0; band < 32; ++band) {
    const int sb = band * 16;
    if (wid == 0) {
      // consume band: 16 (or 15 for the last) sequential recurrence steps
      const float* Pb = sP[band & 1];
      const int smax = (sb + 16 <= NSK_T - 1) ? 16 : (NSK_T - 1 - sb);
      for (int j = 0; j < smax; ++j) {
        float p[16];
#pragma unroll
        for (int i = 0; i < 16; i += 4)
          *(float4*)(p + i) = *(const float4*)(Pb + j * NSK_T + lane * 16 + i);

        // g[i] and local exclusive prefix (register chain, no LDS)
        float exloc[16];
        float run = 0.0f;
#pragma unroll
        for (int i = 0; i < 16; ++i) {
          exloc[i] = run;
          run += fmaf(VARA, prev[i], VARB) * p[i];  // g[511]=0 via zero-padded dY
        }
        const float tot = run;

        // 32-lane inclusive scan of lane totals
        float inc = tot;
#pragma unroll
        for (int off = 1; off < 32; off <<= 1) {
          float nv = __shfl_up(inc, off, 32);
          if (lane >= off) inc += nv;
        }
        const float laneoff = inc - tot;  // exclusive across lanes

#pragma unroll
        for (int i = 0; i < 16; ++i) prev[i] += laneoff + exloc[i];

        float* Orow = Ob + (size_t)(sb + j + 1) * NSK_T + lane * 16;
#pragma unroll
        for (int i = 0; i < 16; i += 4)
          *(float4*)(Orow + i) = *(float4*)(prev + i);
      }
    } else {
      // produce band+1 into the other buffer with waves 1..15
      if (band + 1 < 32) wmma_band((band + 1) * 16, (band + 1) & 1, wid - 1, 15);
    }
    __syncthreads();  // band handoff (uniform across all waves)
  }
}

extern "C" void kernel_launch(void* const* d_in, const int* in_sizes, int n_in,
                              void* d_out, int out_size, void* d_ws, size_t ws_size,
                              hipStream_t stream) {
  (void)in_sizes; (void)n_in; (void)d_ws; (void)ws_size; (void)out_size;
  const float* X = (const float*)d_in[0];
  const float* Y = (const float*)d_in[1];
  float* out = (float*)d_out;
  nsk_kernel<<<dim3(NSK_B), dim3(NSK_T), 0, stream>>>(X, Y, out);
}